// GCN_21251498181438
// MI455X (gfx1250) — compile-verified
//
#include <hip/hip_runtime.h>
#include <hip/hip_bf16.h>

typedef __attribute__((ext_vector_type(2))) float v2f;
typedef __attribute__((ext_vector_type(8))) float v8f;

#define N_NODES 100000
#define N_EDGES 1600000
#define HID 64
#define N_GRAPHS 512
#define HL 256
#define POOL_F 193   // 3*HID + 1

// ---------------------------------------------------------------------------
// Zero-fill (agg buffer must be zeroed before each scatter pass)
// ---------------------------------------------------------------------------
__global__ __launch_bounds__(256) void gnn_zero_f32(float* __restrict__ p, int n) {
    int i = blockIdx.x * 256 + threadIdx.x;
    if (i < n) p[i] = 0.0f;
}

// ---------------------------------------------------------------------------
// Edge scatter-add: agg[dst[e]][f] += h[src[e]][f]
// 64 threads per block = feature dim (coalesced); EPB edges per block.
// h (25.6MB) and agg (25.6MB) both fit in the 192MB L2 -> atomics stay in L2.
// ---------------------------------------------------------------------------
#define EPB 32
__global__ __launch_bounds__(64) void gnn_scatter_add(
    const int* __restrict__ src, const int* __restrict__ dst,
    const float* __restrict__ h, float* __restrict__ agg)
{
    const int f  = threadIdx.x;
    const int e0 = blockIdx.x * EPB;
    int e1 = e0 + EPB;
    if (e1 > N_EDGES) e1 = N_EDGES;
    for (int e = e0; e < e1; ++e) {
        const int s = src[e];
        const int d = dst[e];
        atomicAdd(&agg[d * HID + f], h[s * HID + f]);
    }
}

// ---------------------------------------------------------------------------
// Fused GraphConv GEMM: out = agg @ wrel + brel + h @ wroot  (optional ReLU)
// fp32 WMMA 16x16x4. One wave = one 16x16 output tile. 8 waves/block cover
// 2 M-tiles x 4 N-tiles. Weights staged in LDS (2 * 64*64 * 4B = 32KB).
// ---------------------------------------------------------------------------
__global__ __launch_bounds__(256) void gnn_conv_gemm(
    const float* __restrict__ agg, const float* __restrict__ h,
    const float* __restrict__ wrel, const float* __restrict__ brel,
    const float* __restrict__ wroot, float* __restrict__ out, int do_relu)
{
    __shared__ float sw[2][HID * HID];
    for (int i = threadIdx.x; i < HID * HID; i += 256) {
        sw[0][i] = wrel[i];
        sw[1][i] = wroot[i];
    }
    __syncthreads();

    const int wave = threadIdx.x >> 5;         // 0..7
    const int lane = threadIdx.x & 31;
    const int nt   = wave & 3;                 // N tile (4 tiles of 16 cover HID=64)
    const int mt   = blockIdx.x * 2 + (wave >> 2);  // M tile, 0..6249 (100000 = 16*6250)
    const int arow = mt * 16 + (lane & 15);    // A row for this lane
    const int n    = nt * 16 + (lane & 15);    // output / B column
    const int koff = (lane >> 4) * 2;          // K sub-offset per half-wave

    const float* A0 = agg + arow * HID;
    const float* A1 = h   + arow * HID;

    v8f c = {};
    #pragma unroll
    for (int k0 = 0; k0 < HID; k0 += 4) {
        v2f a, b;
        a.x = A0[k0 + koff + 0];
        a.y = A0[k0 + koff + 1];
        b.x = sw[0][(k0 + koff + 0) * HID + n];
        b.y = sw[0][(k0 + koff + 1) * HID + n];
        c = __builtin_amdgcn_wmma_f32_16x16x4_f32(false, a, false, b, (short)0, c, false, false);
    }
    #pragma unroll
    for (int k0 = 0; k0 < HID; k0 += 4) {
        v2f a, b;
        a.x = A1[k0 + koff + 0];
        a.y = A1[k0 + koff + 1];
        b.x = sw[1][(k0 + koff + 0) * HID + n];
        b.y = sw[1][(k0 + koff + 1) * HID + n];
        c = __builtin_amdgcn_wmma_f32_16x16x4_f32(false, a, false, b, (short)0, c, false, false);
    }

    const float bias = brel[n];
    #pragma unroll
    for (int v = 0; v < 8; ++v) {
        const int r = mt * 16 + v + (lane >> 4) * 8;   // C/D layout: M = v + 8*(lane>=16)
        float val = c[v] + bias;
        if (do_relu) val = fmaxf(val, 0.0f);
        out[r * HID + n] = val;
    }
}

// ---------------------------------------------------------------------------
// Segment pooling: batch is sorted, so block g binary-searches its node range
// and reduces sum/max/count atomic-free. Writes fused MLP input row:
// pooled[g] = [max(64) | mean(64) | sum(64) | T(1)]
// ---------------------------------------------------------------------------
__global__ __launch_bounds__(64) void gnn_pool(
    const float* __restrict__ h, const int* __restrict__ batch,
    const float* __restrict__ T, float* __restrict__ pooled)
{
    __shared__ int se[2];
    const int g = blockIdx.x;
    const int f = threadIdx.x;
    if (f == 0) {
        int lo = 0, hi = N_NODES;
        while (lo < hi) { int mid = (lo + hi) >> 1; if (batch[mid] < g)     lo = mid + 1; else hi = mid; }
        se[0] = lo;
        lo = 0; hi = N_NODES;
        while (lo < hi) { int mid = (lo + hi) >> 1; if (batch[mid] < g + 1) lo = mid + 1; else hi = mid; }
        se[1] = lo;
    }
    __syncthreads();
    const int start = se[0], end = se[1];

    float sum = 0.0f;
    float mx  = -3.4028235e38f;
    for (int nidx = start; nidx < end; ++nidx) {
        const float v = h[nidx * HID + f];
        sum += v;
        mx = fmaxf(mx, v);
    }
    const float cnt  = (float)(end - start);
    const float mean = sum / fmaxf(cnt, 1.0f);
    if (end <= start) mx = 0.0f;   // empty segment: -inf -> 0 (matches reference)

    float* row = pooled + g * POOL_F;
    row[f]            = mx;
    row[HID + f]      = mean;
    row[2 * HID + f]  = sum;
    if (f == 0) row[3 * HID] = T[g];
}

// ---------------------------------------------------------------------------
// MLP hidden layers: one block per graph, 256 threads = output neurons.
// Input row broadcast via LDS.
// ---------------------------------------------------------------------------
__global__ __launch_bounds__(256) void gnn_mlp_layer(
    const float* __restrict__ in, const float* __restrict__ w,
    const float* __restrict__ b, float* __restrict__ out,
    int K, int do_relu)
{
    __shared__ float sin[HL];
    const int g = blockIdx.x;
    const int j = threadIdx.x;
    if (j < K) sin[j] = in[g * K + j];
    __syncthreads();
    float acc = b[j];
    for (int k = 0; k < K; ++k) acc += sin[k] * w[k * HL + j];
    if (do_relu) acc = fmaxf(acc, 0.0f);
    out[g * HL + j] = acc;
}

// Final projection 256 -> 1 per graph (block reduction in LDS)
__global__ __launch_bounds__(256) void gnn_mlp_out(
    const float* __restrict__ in, const float* __restrict__ w3,
    const float* __restrict__ b3, float* __restrict__ out)
{
    __shared__ float red[HL];
    const int g = blockIdx.x;
    const int j = threadIdx.x;
    red[j] = in[g * HL + j] * w3[j];
    __syncthreads();
    for (int s = 128; s > 0; s >>= 1) {
        if (j < s) red[j] += red[j + s];
        __syncthreads();
    }
    if (j == 0) out[g] = red[0] + b3[0];
}

// ---------------------------------------------------------------------------
// Launch
// ---------------------------------------------------------------------------
extern "C" void kernel_launch(void* const* d_in, const int* in_sizes, int n_in,
                              void* d_out, int out_size, void* d_ws, size_t ws_size,
                              hipStream_t stream) {
    (void)in_sizes; (void)n_in; (void)out_size; (void)ws_size;

    const float* x     = (const float*)d_in[0];
    const int*   eidx  = (const int*)d_in[1];
    const int*   batch = (const int*)d_in[2];
    const float* T     = (const float*)d_in[3];
    const float* wrel[4]  = {(const float*)d_in[4],  (const float*)d_in[7],
                             (const float*)d_in[10], (const float*)d_in[13]};
    const float* brel[4]  = {(const float*)d_in[5],  (const float*)d_in[8],
                             (const float*)d_in[11], (const float*)d_in[14]};
    const float* wroot[4] = {(const float*)d_in[6],  (const float*)d_in[9],
                             (const float*)d_in[12], (const float*)d_in[15]};
    const float* w1 = (const float*)d_in[16];
    const float* b1 = (const float*)d_in[17];
    const float* w2 = (const float*)d_in[18];
    const float* b2 = (const float*)d_in[19];
    const float* w3 = (const float*)d_in[20];
    const float* b3 = (const float*)d_in[21];

    const int* src = eidx;            // edge_index[0]
    const int* dst = eidx + N_EDGES;  // edge_index[1]

    // Workspace layout (floats): h_a | h_b | agg | pooled | m1 | m2  (~78.4 MB)
    float* ws     = (float*)d_ws;
    float* h_a    = ws;
    float* h_b    = ws + (size_t)N_NODES * HID;
    float* agg    = ws + (size_t)2 * N_NODES * HID;
    float* pooled = ws + (size_t)3 * N_NODES * HID;
    float* m1     = pooled + N_GRAPHS * HL;   // pooled slot padded to 512*256
    float* m2     = m1 + N_GRAPHS * HL;

    const int nAgg       = N_NODES * HID;                  // 6,400,000
    const int zeroBlocks = (nAgg + 255) / 256;
    const int scatBlocks = (N_EDGES + EPB - 1) / EPB;      // 50,000
    const int gemmBlocks = (N_NODES / 16) / 2;             // 3,125

    const float* h_cur = x;
    float* h_next = h_a;
    for (int L = 0; L < 4; ++L) {
        gnn_zero_f32<<<zeroBlocks, 256, 0, stream>>>(agg, nAgg);
        gnn_scatter_add<<<scatBlocks, 64, 0, stream>>>(src, dst, h_cur, agg);
        gnn_conv_gemm<<<gemmBlocks, 256, 0, stream>>>(
            agg, h_cur, wrel[L], brel[L], wroot[L], h_next, (L < 3) ? 1 : 0);
        h_cur  = h_next;
        h_next = (h_next == h_a) ? h_b : h_a;
    }
    // h_cur now points at the layer-3 output (h_b)

    gnn_pool<<<N_GRAPHS, 64, 0, stream>>>(h_cur, batch, T, pooled);
    gnn_mlp_layer<<<N_GRAPHS, HL, 0, stream>>>(pooled, w1, b1, m1, POOL_F, 1);
    gnn_mlp_layer<<<N_GRAPHS, HL, 0, stream>>>(m1, w2, b2, m2, HL, 1);
    gnn_mlp_out<<<N_GRAPHS, HL, 0, stream>>>(m2, w3, b3, (float*)d_out);
}